// ProductKGE_7576322310252
// MI455X (gfx1250) — compile-verified
//
#include <hip/hip_runtime.h>
#include <math.h>

// ProductKGE scoring for MI455X (gfx1250, wave32).
// Memory-bound gather (~400 B/sample); the three 16x16 f32 transforms are put
// on the WMMA pipe (v_wmma_f32_16x16x4_f32, 4 chained K-chunks each) so the
// VALU only handles projections/distances. One wave = 16 samples:
//   Dt = W (M=j,K=k) x H^T (K=k,N=s)  => lane L owns sample s=L&15,
//   output dims j in [8*(L>>4), 8*(L>>4)+7] across the 8 acc VGPRs.
// 16-dim reductions = 8-wide in-lane sum + one shfl_xor(16) with partner lane.

typedef __attribute__((ext_vector_type(2))) float v2f;
typedef __attribute__((ext_vector_type(8))) float v8f;

#define NDIM 48
#define EPSF 1e-5f

__device__ __forceinline__ float pair_sum(float x) {
    // full 16-dim sum: this lane's 8 elements + partner lane (L ^ 16)
    return x + __shfl_xor(x, 16, 32);
}

__device__ __forceinline__ v8f mm16x16(const v2f a[4], const v2f b[4]) {
    v8f c = {};
#pragma unroll
    for (int q = 0; q < 4; ++q) {
        // D = A(16x4 f32) x B(4x16 f32) + C, accumulated over 4 K-chunks
        c = __builtin_amdgcn_wmma_f32_16x16x4_f32(false, a[q], false, b[q],
                                                  (short)0, c, false, false);
    }
    return c;
}

__global__ __launch_bounds__(256)
void productkge_wmma(const float* __restrict__ emb,
                     const float* __restrict__ Wh, const float* __restrict__ bh,
                     const float* __restrict__ Ws, const float* __restrict__ bs,
                     const float* __restrict__ We, const float* __restrict__ be,
                     const int* __restrict__ heads, const int* __restrict__ rels,
                     const int* __restrict__ tails, float* __restrict__ out)
{
    const int lane = threadIdx.x & 31;
    const int wave = threadIdx.x >> 5;
    const int hh   = lane >> 4;      // lane half: selects K/J sub-range
    const int m    = lane & 15;      // sample-in-tile == matrix column N
    const int sample = (blockIdx.x * 8 + wave) * 16 + m;

    const int hidx = heads[sample];
    const int tidx = tails[sample];
    const int rel  = rels[sample];
    const float* hrow = emb + (size_t)hidx * NDIM;
    const float* trow = emb + (size_t)tidx * NDIM;

    // ---- A fragments: A[j][k] = W[j][k]; lane holds M=j=m, K=4q+2hh+{0,1} ----
    // ---- B fragments: B[k][s] = H[s][k]; lane holds N=s=m, K=4q+2hh+{0,1} ----
    v2f aH[4], aS[4], aE[4], bHy[4], bSp[4], bEu[4];
#pragma unroll
    for (int q = 0; q < 4; ++q) {
        const int wo = m * 16 + 4 * q + 2 * hh;
        aH[q] = *(const v2f*)(Wh + wo);
        aS[q] = *(const v2f*)(Ws + wo);
        aE[q] = *(const v2f*)(We + wo);
        const int ho = 4 * q + 2 * hh;
        bHy[q] = *(const v2f*)(hrow + 0  + ho);
        bSp[q] = *(const v2f*)(hrow + 16 + ho);
        bEu[q] = *(const v2f*)(hrow + 32 + ho);
    }

    // ---- project head hyp (norm-clip) and sph (normalize) segments ----
    float sh = 0.f, ss = 0.f;
#pragma unroll
    for (int q = 0; q < 4; ++q) {
        sh += bHy[q].x * bHy[q].x + bHy[q].y * bHy[q].y;
        ss += bSp[q].x * bSp[q].x + bSp[q].y * bSp[q].y;
    }
    const float hn      = sqrtf(pair_sum(sh));
    const float h_scale = (hn > 1.f - EPSF) ? (1.f - EPSF) / fmaxf(hn, EPSF) : 1.f;
    const float s_scale = 1.f / fmaxf(sqrtf(pair_sum(ss)), 1e-7f);
#pragma unroll
    for (int q = 0; q < 4; ++q) { bHy[q] *= h_scale; bSp[q] *= s_scale; }

    float u[8], tv[8];
    float d_hyp, d_sph, d_euc;

    // =================== hyperbolic (Poincare) ===================
    {
        v8f acc = mm16x16(aH, bHy);
        const float4 b0 = *(const float4*)(bh + 8 * hh);
        const float4 b1 = *(const float4*)(bh + 8 * hh + 4);
        u[0]=acc[0]+b0.x; u[1]=acc[1]+b0.y; u[2]=acc[2]+b0.z; u[3]=acc[3]+b0.w;
        u[4]=acc[4]+b1.x; u[5]=acc[5]+b1.y; u[6]=acc[6]+b1.z; u[7]=acc[7]+b1.w;
        const float4 t0 = *(const float4*)(trow + 0 + 8 * hh);
        const float4 t1 = *(const float4*)(trow + 0 + 8 * hh + 4);
        tv[0]=t0.x; tv[1]=t0.y; tv[2]=t0.z; tv[3]=t0.w;
        tv[4]=t1.x; tv[5]=t1.y; tv[6]=t1.z; tv[7]=t1.w;

        float tssq = 0.f, ussq = 0.f;
#pragma unroll
        for (int r = 0; r < 8; ++r) { tssq += tv[r]*tv[r]; ussq += u[r]*u[r]; }
        const float ttot = pair_sum(tssq);
        const float utot = pair_sum(ussq);
        const float tn = sqrtf(ttot), unr = sqrtf(utot);
        const float st = (tn  > 1.f - EPSF) ? (1.f - EPSF) / fmaxf(tn,  EPSF) : 1.f;
        const float su = (unr > 1.f - EPSF) ? (1.f - EPSF) / fmaxf(unr, EPSF) : 1.f;
        const float vn = st * st * ttot;
        const float un = su * su * utot;
        float dsq = 0.f;
#pragma unroll
        for (int r = 0; r < 8; ++r) { float d = su*u[r] - st*tv[r]; dsq += d*d; }
        const float duv = pair_sum(dsq);
        const float arg = 1.f + 2.f * duv / fmaxf((1.f - un) * (1.f - vn), EPSF);
        d_hyp = acoshf(fmaxf(arg, 1.f + EPSF));
    }

    // =================== spherical ===================
    {
        v8f acc = mm16x16(aS, bSp);
        const float4 b0 = *(const float4*)(bs + 8 * hh);
        const float4 b1 = *(const float4*)(bs + 8 * hh + 4);
        u[0]=acc[0]+b0.x; u[1]=acc[1]+b0.y; u[2]=acc[2]+b0.z; u[3]=acc[3]+b0.w;
        u[4]=acc[4]+b1.x; u[5]=acc[5]+b1.y; u[6]=acc[6]+b1.z; u[7]=acc[7]+b1.w;
        const float4 t0 = *(const float4*)(trow + 16 + 8 * hh);
        const float4 t1 = *(const float4*)(trow + 16 + 8 * hh + 4);
        tv[0]=t0.x; tv[1]=t0.y; tv[2]=t0.z; tv[3]=t0.w;
        tv[4]=t1.x; tv[5]=t1.y; tv[6]=t1.z; tv[7]=t1.w;

        float tssq = 0.f, ussq = 0.f;
#pragma unroll
        for (int r = 0; r < 8; ++r) { tssq += tv[r]*tv[r]; ussq += u[r]*u[r]; }
        const float st = 1.f / fmaxf(sqrtf(pair_sum(tssq)), 1e-7f);
        const float su = 1.f / fmaxf(sqrtf(pair_sum(ussq)), 1e-7f);
        float dp = 0.f;
#pragma unroll
        for (int r = 0; r < 8; ++r) dp += (su * u[r]) * (st * tv[r]);
        float dot = pair_sum(dp);
        dot = fminf(fmaxf(dot, -1.f + EPSF), 1.f - EPSF);
        d_sph = acosf(dot);
    }

    // =================== Euclidean ===================
    {
        v8f acc = mm16x16(aE, bEu);
        const float4 b0 = *(const float4*)(be + 8 * hh);
        const float4 b1 = *(const float4*)(be + 8 * hh + 4);
        u[0]=acc[0]+b0.x; u[1]=acc[1]+b0.y; u[2]=acc[2]+b0.z; u[3]=acc[3]+b0.w;
        u[4]=acc[4]+b1.x; u[5]=acc[5]+b1.y; u[6]=acc[6]+b1.z; u[7]=acc[7]+b1.w;
        const float4 t0 = *(const float4*)(trow + 32 + 8 * hh);
        const float4 t1 = *(const float4*)(trow + 32 + 8 * hh + 4);
        tv[0]=t0.x; tv[1]=t0.y; tv[2]=t0.z; tv[3]=t0.w;
        tv[4]=t1.x; tv[5]=t1.y; tv[6]=t1.z; tv[7]=t1.w;
        float dsq = 0.f;
#pragma unroll
        for (int r = 0; r < 8; ++r) { float d = u[r] - tv[r]; dsq += d*d; }
        d_euc = sqrtf(pair_sum(dsq));
    }

    const float d = (rel == 0) ? d_hyp : (rel == 1) ? d_sph : d_euc;
    if (hh == 0) out[sample] = -d;
}

// ---------------- scalar fallback for a non-multiple-of-128 tail ----------------
__device__ __forceinline__ void proj_hyp16(float* v) {
    float s = 0.f;
    for (int k = 0; k < 16; ++k) s += v[k] * v[k];
    const float n = sqrtf(s);
    const float sc = (n > 1.f - EPSF) ? (1.f - EPSF) / fmaxf(n, EPSF) : 1.f;
    for (int k = 0; k < 16; ++k) v[k] *= sc;
}
__device__ __forceinline__ void proj_sph16(float* v) {
    float s = 0.f;
    for (int k = 0; k < 16; ++k) s += v[k] * v[k];
    const float sc = 1.f / fmaxf(sqrtf(s), 1e-7f);
    for (int k = 0; k < 16; ++k) v[k] *= sc;
}

__global__ void productkge_scalar(const float* __restrict__ emb,
                                  const float* __restrict__ Wh, const float* __restrict__ bh,
                                  const float* __restrict__ Ws, const float* __restrict__ bs,
                                  const float* __restrict__ We, const float* __restrict__ be,
                                  const int* __restrict__ heads, const int* __restrict__ rels,
                                  const int* __restrict__ tails, float* __restrict__ out,
                                  int base, int nB)
{
    const int i = base + blockIdx.x * blockDim.x + threadIdx.x;
    if (i >= nB) return;
    const float* hr = emb + (size_t)heads[i] * NDIM;
    const float* tr = emb + (size_t)tails[i] * NDIM;
    float hH[16], hS[16], hE[16], tH[16], tS[16], tE[16];
    for (int k = 0; k < 16; ++k) {
        hH[k] = hr[k]; hS[k] = hr[16 + k]; hE[k] = hr[32 + k];
        tH[k] = tr[k]; tS[k] = tr[16 + k]; tE[k] = tr[32 + k];
    }
    proj_hyp16(hH); proj_sph16(hS); proj_hyp16(tH); proj_sph16(tS);
    float uH[16], uS[16], uE[16];
    for (int j = 0; j < 16; ++j) {
        float a = bh[j], b = bs[j], c = be[j];
        for (int k = 0; k < 16; ++k) {
            a += Wh[j * 16 + k] * hH[k];
            b += Ws[j * 16 + k] * hS[k];
            c += We[j * 16 + k] * hE[k];
        }
        uH[j] = a; uS[j] = b; uE[j] = c;
    }
    proj_hyp16(uH); proj_sph16(uS);
    float un = 0.f, vn = 0.f, duv = 0.f, dot = 0.f, esq = 0.f;
    for (int k = 0; k < 16; ++k) {
        un += uH[k] * uH[k]; vn += tH[k] * tH[k];
        float d = uH[k] - tH[k]; duv += d * d;
        dot += uS[k] * tS[k];
        float e = uE[k] - tE[k]; esq += e * e;
    }
    const float arg = 1.f + 2.f * duv / fmaxf((1.f - un) * (1.f - vn), EPSF);
    const float dh = acoshf(fmaxf(arg, 1.f + EPSF));
    const float ds = acosf(fminf(fmaxf(dot, -1.f + EPSF), 1.f - EPSF));
    const float de = sqrtf(esq);
    const int r = rels[i];
    out[i] = -((r == 0) ? dh : (r == 1) ? ds : de);
}

extern "C" void kernel_launch(void* const* d_in, const int* in_sizes, int n_in,
                              void* d_out, int out_size, void* d_ws, size_t ws_size,
                              hipStream_t stream) {
    const float* emb = (const float*)d_in[0];
    const float* Wh  = (const float*)d_in[1];
    const float* bh  = (const float*)d_in[2];
    const float* Ws  = (const float*)d_in[3];
    const float* bs  = (const float*)d_in[4];
    const float* We  = (const float*)d_in[5];
    const float* be  = (const float*)d_in[6];
    const int* heads = (const int*)d_in[7];
    const int* rels  = (const int*)d_in[8];
    const int* tails = (const int*)d_in[9];
    float* out = (float*)d_out;

    const int nB   = in_sizes[7];        // batch size (heads count)
    const int full = nB / 128;           // 128 samples per 256-thread block (8 waves x 16)
    if (full > 0) {
        productkge_wmma<<<full, 256, 0, stream>>>(emb, Wh, bh, Ws, bs, We, be,
                                                  heads, rels, tails, out);
    }
    const int rem = nB - full * 128;
    if (rem > 0) {
        productkge_scalar<<<(rem + 63) / 64, 64, 0, stream>>>(
            emb, Wh, bh, Ws, bs, We, be, heads, rels, tails, out, full * 128, nB);
    }
    (void)n_in; (void)out_size; (void)d_ws; (void)ws_size;
}